// QLSTMHybridQuantum_65481071406312
// MI455X (gfx1250) — compile-verified
//
#include <hip/hip_runtime.h>
#include <math.h>

// ---------------------------------------------------------------------------
// QLSTM hybrid for MI455X (gfx1250, wave32).
//
// Exact algebraic collapse of the quantum circuit:
//   * RX(x)RX(th) = RX(x+th); product state => P(bit_w=1)=sin^2(a_w/2).
//   * CNOT chain = XOR-linear permutation of basis states:
//       final bit 0 = b1^...^b7 ; final bit k = b0^...^bk (k>=1)
//   * E[(-1)^{xor of S}] over a product distribution = prod_{v in S} cos(a_v)
//   => gate_layer(x)[0]  = prod_{v=1..7} cos(x_v+th_v)
//      gate_layer(x)[k]  = prod_{v=0..k} cos(x_v+th_v)   (prefix products)
//
// Pipeline:
//   K1: Xpre[2048][32] = gather(emb) @ [Wf;Wi;Wu;Wo][:, :256]^T + (bias+theta)
//       via V_WMMA_F32_16X16X4_F32 (exact f32 tensor path).
//   K2: single-wave sequential scan (2048 steps): registers + shuffles only,
//       hardware transcendentals (v_cos/v_exp/v_rcp) on the critical chain.
//   K3: logits = outs @ Wt^T + bt, log_softmax; one wave per timestep.
// ---------------------------------------------------------------------------

typedef __attribute__((ext_vector_type(2))) float v2f;
typedef __attribute__((ext_vector_type(8))) float v8f;

#define SEQL 2048
#define EMBD 256
#define HIDD 8
#define NTAG 50
#define DIN  264   // EMB + HID

// ---- fast device math (single-instruction transcendentals) ----------------
__device__ __forceinline__ float fast_rcp(float x) { return __builtin_amdgcn_rcpf(x); }
__device__ __forceinline__ float fast_sigmoid(float z) {
    return fast_rcp(1.0f + __expf(-z));          // v_exp_f32 + v_rcp_f32
}
__device__ __forceinline__ float fast_tanh(float x) {
    const float a = fabsf(x);
    const float e = __expf(-2.0f * a);           // in (0,1], never overflows
    const float r = (1.0f - e) * fast_rcp(1.0f + e);
    return copysignf(r, x);
}

// ---------------- K1: embedding gather + x-projection GEMM (WMMA f32) ------
__global__ __launch_bounds__(32) void xproj_kernel(
    const int*   __restrict__ sentence,
    const float* __restrict__ emb,
    const float* __restrict__ Wf, const float* __restrict__ bf,
    const float* __restrict__ Wi, const float* __restrict__ bi,
    const float* __restrict__ Wu, const float* __restrict__ bu,
    const float* __restrict__ Wo, const float* __restrict__ bo,
    const float* __restrict__ thf, const float* __restrict__ thi,
    const float* __restrict__ thu, const float* __restrict__ tho,
    float* __restrict__ Xpre)
{
    const int lane  = threadIdx.x;
    const int mtile = blockIdx.x >> 1;   // 128 tiles of 16 timesteps
    const int ntile = blockIdx.x & 1;    // 2 tiles of 16 gate-outputs
    const int half  = lane >> 4;         // 0: K+{0,1}, 1: K+{2,3} (f32 A/B layout)
    const int l15   = lane & 15;

    // A fragment source: row m = timestep, elements emb[tok[m]][k]
    const int m   = mtile * 16 + l15;
    const int tok = sentence[m];
    const float* arow = emb + (size_t)tok * EMBD;

    // B fragment source: col n = gate output (g*8+q), elements Wg[q][k]
    const int n = ntile * 16 + l15;
    const int g = n >> 3, q = n & 7;
    const float* Wg = (g == 0) ? Wf  : (g == 1) ? Wi  : (g == 2) ? Wu  : Wo;
    const float* bg = (g == 0) ? bf  : (g == 1) ? bi  : (g == 2) ? bu  : bo;
    const float* tg = (g == 0) ? thf : (g == 1) ? thi : (g == 2) ? thu : tho;
    const float* brow = Wg + q * DIN;
    const float  bias = bg[q] + tg[q];   // fold quantum rotation offset here

    v8f c = {};
    #pragma unroll 4
    for (int kk = 0; kk < EMBD; kk += 4) {
        const int k0 = kk + 2 * half;
        v2f a, b;
        a.x = arow[k0]; a.y = arow[k0 + 1];
        b.x = brow[k0]; b.y = brow[k0 + 1];
        c = __builtin_amdgcn_wmma_f32_16x16x4_f32(
                /*neg_a=*/false, a, /*neg_b=*/false, b,
                /*c_mod=*/(short)0, c, /*reuse_a=*/false, /*reuse_b=*/false);
    }

    // C layout: VGPR r -> M = r (lanes 0-15) / r+8 (lanes 16-31), N = lane&15
    const int col = ntile * 16 + l15;
    #pragma unroll
    for (int r = 0; r < 8; ++r) {
        const int row = mtile * 16 + r + 8 * half;
        Xpre[row * 32 + col] = c[r] + bias;
    }
}

// ---------------- K2: single-wave sequential LSTM scan ---------------------
__global__ __launch_bounds__(32) void scan_kernel(
    const float* __restrict__ Xpre,
    const float* __restrict__ Wf, const float* __restrict__ Wi,
    const float* __restrict__ Wu, const float* __restrict__ Wo,
    float* __restrict__ outs)
{
    const int lane = threadIdx.x;        // 32 lanes = 4 gates x 8 wires
    const int g = lane >> 3, q = lane & 7;

    const float* Wg = (g == 0) ? Wf : (g == 1) ? Wi : (g == 2) ? Wu : Wo;
    float wh[HIDD];
    #pragma unroll
    for (int k = 0; k < HIDD; ++k) wh[k] = Wg[q * DIN + EMBD + k]; // h-part of W

    float hx = 0.0f;   // lanes 0..7 hold h[q]
    float cx = 0.0f;   // lanes 0..7 hold c[q]

    float xp = Xpre[lane];               // software-pipelined Xpre load
    for (int t = 0; t < SEQL; ++t) {
        const float xpn = (t + 1 < SEQL) ? Xpre[(t + 1) * 32 + lane] : 0.0f;

        // wire angle: hoisted (x-proj + bias + theta) + recurrent h-part
        float a = xp;
        #pragma unroll
        for (int k = 0; k < HIDD; ++k)
            a += __shfl(hx, k, 32) * wh[k];

        const float cw = __cosf(a);      // v_cos_f32

        // inclusive prefix product of cos over the 8-lane group (wires 0..q)
        float p = cw;
        #pragma unroll
        for (int s = 1; s < 8; s <<= 1) {
            const float v = __shfl(p, (q >= s) ? (lane - s) : lane, 32);
            if (q >= s) p *= v;
        }
        // product over wires 1..7 (output for wire 0); runs parallel to scan
        float e = (q == 0) ? 1.0f : cw;
        #pragma unroll
        for (int s = 1; s < 8; s <<= 1) e *= __shfl_xor(e, s, 32);

        const float z   = (q == 0) ? e : p;
        const float act = (g == 2) ? fast_tanh(z) : fast_sigmoid(z);

        // gather f,i,u,o for wire q; update cell (same result in 4 lane groups)
        const float fq = __shfl(act, q,      32);
        const float iq = __shfl(act, 8 + q,  32);
        const float uq = __shfl(act, 16 + q, 32);
        const float oq = __shfl(act, 24 + q, 32);
        cx = fq * cx + iq * uq;
        hx = oq * fast_tanh(cx);

        if (lane < HIDD) outs[t * HIDD + q] = hx;
        xp = xpn;
    }
}

// ---------------- K3: output projection + log_softmax ----------------------
__global__ __launch_bounds__(256) void logits_kernel(
    const float* __restrict__ outs,
    const float* __restrict__ Wt, const float* __restrict__ bt,
    float* __restrict__ out)
{
    const int wave = threadIdx.x >> 5;
    const int lane = threadIdx.x & 31;
    const int t = blockIdx.x * 8 + wave;   // one wave per timestep

    float hv[HIDD];
    #pragma unroll
    for (int k = 0; k < HIDD; ++k) hv[k] = outs[t * HIDD + k];

    // each lane handles tags {lane, lane+32}
    float la = bt[lane];
    #pragma unroll
    for (int k = 0; k < HIDD; ++k) la += Wt[lane * HIDD + k] * hv[k];

    const bool has2 = (lane + 32) < NTAG;
    float lb = -INFINITY;
    if (has2) {
        lb = bt[lane + 32];
        #pragma unroll
        for (int k = 0; k < HIDD; ++k) lb += Wt[(lane + 32) * HIDD + k] * hv[k];
    }

    float m = has2 ? fmaxf(la, lb) : la;
    #pragma unroll
    for (int s = 16; s; s >>= 1) m = fmaxf(m, __shfl_xor(m, s, 32));

    float ssum = __expf(la - m) + (has2 ? __expf(lb - m) : 0.0f);
    #pragma unroll
    for (int s = 16; s; s >>= 1) ssum += __shfl_xor(ssum, s, 32);
    const float ls = __logf(ssum);

    out[t * NTAG + lane] = la - m - ls;
    if (has2) out[t * NTAG + lane + 32] = lb - m - ls;
}

// ---------------------------------------------------------------------------
extern "C" void kernel_launch(void* const* d_in, const int* in_sizes, int n_in,
                              void* d_out, int out_size, void* d_ws, size_t ws_size,
                              hipStream_t stream)
{
    const int*   sentence = (const int*)  d_in[0];
    const float* emb      = (const float*)d_in[1];
    const float* Wf       = (const float*)d_in[2];
    const float* bf       = (const float*)d_in[3];
    const float* Wi       = (const float*)d_in[4];
    const float* bi       = (const float*)d_in[5];
    const float* Wu       = (const float*)d_in[6];
    const float* bu       = (const float*)d_in[7];
    const float* Wo       = (const float*)d_in[8];
    const float* bo       = (const float*)d_in[9];
    const float* thf      = (const float*)d_in[10];
    const float* thi      = (const float*)d_in[11];
    const float* thu      = (const float*)d_in[12];
    const float* tho      = (const float*)d_in[13];
    const float* Wt       = (const float*)d_in[14];
    const float* bt       = (const float*)d_in[15];

    float* Xpre = (float*)d_ws;              // 2048*32 f32 = 256 KB
    float* outs = Xpre + SEQL * 32;          // 2048*8  f32 =  64 KB

    xproj_kernel<<<256, 32, 0, stream>>>(sentence, emb, Wf, bf, Wi, bi,
                                         Wu, bu, Wo, bo,
                                         thf, thi, thu, tho, Xpre);
    scan_kernel<<<1, 32, 0, stream>>>(Xpre, Wf, Wi, Wu, Wo, outs);
    logits_kernel<<<SEQL / 8, 256, 0, stream>>>(outs, Wt, bt, (float*)d_out);
}